// GCNNet_62508954026414
// MI455X (gfx1250) — compile-verified
//
#include <hip/hip_runtime.h>
#include <hip/hip_bf16.h>
#include <math.h>

typedef __attribute__((ext_vector_type(2))) float v2f;
typedef __attribute__((ext_vector_type(8))) float v8f;

#define F_IN 512
#define H    16
#define C    40

// ---------------- degree / norm ----------------
__global__ void deg_init_kernel(float* deg, int N) {
    int i = blockIdx.x * blockDim.x + threadIdx.x;
    if (i < N) deg[i] = 1.0f;                 // self-loop contributes 1
}

__global__ void deg_edge_kernel(const long long* __restrict__ ei, float* deg, int E) {
    int e = blockIdx.x * blockDim.x + threadIdx.x;
    if (e < E) {
        long long d = ei[E + e];              // dst row
        atomicAdd(&deg[(int)d], 1.0f);
    }
}

__global__ void dinv_kernel(float* deg, int N) {
    int i = blockIdx.x * blockDim.x + threadIdx.x;
    if (i < N) deg[i] = rsqrtf(deg[i]);       // in place: deg -> deg^-1/2 (deg >= 1)
}

// ---------------- layer-1 GEMM: h1 = X @ W1 via v_wmma_f32_16x16x4_f32 ----------------
__global__ __launch_bounds__(256) void gemm1_wmma_kernel(const float* __restrict__ x,
                                                         const float* __restrict__ W1,
                                                         float* __restrict__ h1, int N) {
    __shared__ float sW[F_IN * H];            // 32 KB (of 320 KB/WGP)
    for (int i = threadIdx.x; i < F_IN * H; i += blockDim.x) sW[i] = W1[i];
    __syncthreads();

    const int wave = threadIdx.x >> 5;        // 8 waves per block
    const int lane = threadIdx.x & 31;
    const int tile = blockIdx.x * 8 + wave;   // one 16-row M tile per wave
    const int row0 = tile * 16;
    if (row0 >= N) return;

    const int r    = lane & 15;               // A: M row / B: N col / D: N col
    const int koff = (lane >> 4) << 1;        // lanes 16-31 hold K+2 (ISA 7.12.2)
    const float* xrow = x + (size_t)(row0 + r) * F_IN + koff;

    v8f c = {};
    for (int k = 0; k < F_IN; k += 4) {
        v2f a, b;
        a.x = xrow[k];                        // A[M=r][K=k+koff]
        a.y = xrow[k + 1];                    // A[M=r][K=k+koff+1]
        b.x = sW[(k + koff) * H + r];         // B[K][N=r]
        b.y = sW[(k + koff + 1) * H + r];
        // D = A(16x4 f32) * B(4x16 f32) + C   -> v_wmma_f32_16x16x4_f32
        c = __builtin_amdgcn_wmma_f32_16x16x4_f32(
                /*neg_a=*/false, a, /*neg_b=*/false, b,
                /*c_mod=*/(short)0, c, /*reuse_a=*/false, /*reuse_b=*/false);
    }
    // D layout: VGPR v -> row row0 + v + 8*(lane>=16), col r
    float* out = h1 + (size_t)(row0 + ((lane >> 4) << 3)) * H + r;
#pragma unroll
    for (int v = 0; v < 8; ++v) out[v * H] = c[v];
}

// ---------------- self-loop init: agg[i] = h[i] * dinv[i]^2 ----------------
__global__ void self_loop_kernel(const float* __restrict__ h, const float* __restrict__ dinv,
                                 float* __restrict__ agg, int N) {
    int idx = blockIdx.x * blockDim.x + threadIdx.x;
    if (idx < N * H) {
        int i = idx >> 4;
        float s = dinv[i];
        agg[idx] = h[idx] * s * s;
    }
}

// ---------------- edge scatter: agg[dst] += h[src] * dinv[src]*dinv[dst] ----------------
__global__ void edge_agg_kernel(const long long* __restrict__ ei,
                                const float* __restrict__ h,
                                const float* __restrict__ dinv,
                                float* __restrict__ agg, int E) {
    int tid = blockIdx.x * blockDim.x + threadIdx.x;   // (edge, feature) pair
    if (tid >= E * H) return;
    int e = tid >> 4;
    int f = tid & 15;
    int s = (int)ei[e];
    int d = (int)ei[E + e];
    float nrm = dinv[s] * dinv[d];
    atomicAdd(&agg[(size_t)d * H + f], h[(size_t)s * H + f] * nrm);
}

// ---------------- bias + relu (in place) ----------------
__global__ void bias_relu_kernel(float* agg, const float* __restrict__ b1, int N) {
    int idx = blockIdx.x * blockDim.x + threadIdx.x;
    if (idx < N * H) {
        float v = agg[idx] + b1[idx & 15];
        agg[idx] = v > 0.0f ? v : 0.0f;
    }
}

// ---------------- fused: out = log_softmax(agg2 @ W2 + b2) ----------------
__global__ __launch_bounds__(256) void out_kernel(const float* __restrict__ agg2,
                                                  const float* __restrict__ W2,
                                                  const float* __restrict__ b2,
                                                  float* __restrict__ out, int N) {
    __shared__ float sW[H * C];
    __shared__ float sb[C];
    for (int i = threadIdx.x; i < H * C; i += blockDim.x) sW[i] = W2[i];
    if (threadIdx.x < C) sb[threadIdx.x] = b2[threadIdx.x];
    __syncthreads();

    int i = blockIdx.x * blockDim.x + threadIdx.x;
    if (i >= N) return;

    float hrow[H];
    const float* row = agg2 + (size_t)i * H;
#pragma unroll
    for (int k = 0; k < H; ++k) hrow[k] = row[k];

    float o[C];
#pragma unroll
    for (int c = 0; c < C; ++c) {
        float acc = sb[c];
#pragma unroll
        for (int k = 0; k < H; ++k) acc = fmaf(hrow[k], sW[k * C + c], acc);
        o[c] = acc;
    }
    float m = o[0];
#pragma unroll
    for (int c = 1; c < C; ++c) m = fmaxf(m, o[c]);
    float s = 0.0f;
#pragma unroll
    for (int c = 0; c < C; ++c) s += expf(o[c] - m);
    float lse = m + logf(s);
    float* orow = out + (size_t)i * C;
#pragma unroll
    for (int c = 0; c < C; ++c) orow[c] = o[c] - lse;
}

extern "C" void kernel_launch(void* const* d_in, const int* in_sizes, int n_in,
                              void* d_out, int out_size, void* d_ws, size_t ws_size,
                              hipStream_t stream) {
    const float*      x  = (const float*)d_in[0];
    const long long*  ei = (const long long*)d_in[1];   // int64 [2, E]
    const float*      W1 = (const float*)d_in[2];       // [512,16]
    const float*      b1 = (const float*)d_in[3];       // [16]
    const float*      W2 = (const float*)d_in[4];       // [16,40]
    const float*      b2 = (const float*)d_in[5];       // [40]
    float*            out = (float*)d_out;

    const int N = in_sizes[0] / F_IN;   // 100000
    const int E = in_sizes[1] / 2;      // 3200000

    float* ws   = (float*)d_ws;
    float* dinv = ws;                   // N      (deg, then dinv in place)
    float* h1   = ws + N;               // 16N    X@W1
    float* agg1 = ws + (size_t)17 * N;  // 16N    layer-1 aggregate / relu_h (in place)
    float* agg2 = ws + (size_t)33 * N;  // 16N    layer-2 aggregate (pre-W2)

    const int TB = 256;
    const int nbN   = (N + TB - 1) / TB;
    const int nbE   = (E + TB - 1) / TB;
    const int nbNH  = (N * H + TB - 1) / TB;
    const int nbEH  = (E * H + TB - 1) / TB;
    const int nbGemm = ((N + 15) / 16 + 7) / 8;         // 8 M-tiles (waves) per block

    // deg -> dinv
    deg_init_kernel<<<nbN, TB, 0, stream>>>(dinv, N);
    deg_edge_kernel<<<nbE, TB, 0, stream>>>(ei, dinv, E);
    dinv_kernel<<<nbN, TB, 0, stream>>>(dinv, N);

    // layer 1: h1 = X @ W1  (WMMA), then Â-aggregate, bias+relu
    gemm1_wmma_kernel<<<nbGemm, TB, 0, stream>>>(x, W1, h1, N);
    self_loop_kernel<<<nbNH, TB, 0, stream>>>(h1, dinv, agg1, N);
    edge_agg_kernel<<<nbEH, TB, 0, stream>>>(ei, h1, dinv, agg1, E);
    bias_relu_kernel<<<nbNH, TB, 0, stream>>>(agg1, b1, N);

    // layer 2: aggregate FIRST (W2 commutes with segment-sum), 16 feats/edge not 40
    self_loop_kernel<<<nbNH, TB, 0, stream>>>(agg1, dinv, agg2, N);
    edge_agg_kernel<<<nbEH, TB, 0, stream>>>(ei, agg1, dinv, agg2, E);

    // out = log_softmax(agg2 @ W2 + b2)
    out_kernel<<<nbN, TB, 0, stream>>>(agg2, W2, b2, out, N);
}